// RefineHead_74938589381108
// MI455X (gfx1250) — compile-verified
//
#include <hip/hip_runtime.h>

typedef __attribute__((ext_vector_type(2))) float v2f;
typedef __attribute__((ext_vector_type(8))) float v8f;

#if defined(__has_builtin)
#if __has_builtin(__builtin_amdgcn_wmma_f32_16x16x4_f32)
#define HAVE_WMMA_F32X4 1
#endif
#endif

// ---------------- problem constants ----------------
constexpr int kB = 8, kP = 768, kS = 36, kCin = 64, kC = 72, kD = 512;
constexpr int kNS = 71;    // N_STRIPS
constexpr int kPr = 76;    // priors row length (4 + 72)
constexpr int kOut = 78;   // output row length (2 + 4 + 72)

// ---------------- workspace layout (float offsets) ----------------
constexpr long long OFF_FEATS0 = 0;          // 8*4000*72  = 2,304,000
constexpr long long OFF_FEATS1 = 2304000;    // 8*1000*72  =   576,000
constexpr long long OFF_FEATS2 = 2880000;    // 8* 250*72  =   144,000
constexpr long long OFF_OFFS0  = 3024000;    // 2,304,000
constexpr long long OFF_OFFS1  = 5328000;    //   576,000
constexpr long long OFF_OFFS2  = 5904000;    //   144,000
constexpr long long OFF_COMB   = 6048000;    // 8*768*36*72 = 15,925,248
constexpr long long OFF_WF     = 21973248;   // 512*9*72    =    331,776
constexpr long long OFF_BF     = 22305024;   // 512
constexpr long long OFF_FEAT   = 22305536;   // 6144*512    =  3,145,728
constexpr long long OFF_Q      = 25451264;
constexpr long long OFF_K      = 28596992;
constexpr long long OFF_V      = 31742720;
constexpr long long OFF_SCORES = 0;          // 32*768*768 = 18,874,368 (reuses dead sampling bufs)
constexpr long long OFF_RES    = 34888448;   // 3,145,728
constexpr long long OFF_RES2   = 25451264;   // reuses Q
constexpr long long OFF_T1     = 28596992;   // 6,291,456 reuses K+V
constexpr long long OFF_FEAT2  = 0;          // reuses scores
constexpr long long OFF_CLSH   = 3145728;
constexpr long long OFF_REG    = 6291456;
constexpr long long OFF_PH     = 9437184;    // 6,291,456
constexpr long long OFF_COEF   = 15728640;   // 6144*4
// total = 38,034,176 floats = ~152 MB

#define FLAG_BIAS 1
#define FLAG_RELU 2
#define FLAG_ADD  4
#define FLAG_SCALE 8

// ================= generic fp32 WMMA GEMM, 64x32 per wave =================
// C[M,N] = epi( alpha * A[M,K] x B[K,N] + bias )  with
//   !transB : B[k,n] = W[n*ldw + k]   (W row-major [N,K], the x @ w.T case)
//    transB : B[k,n] = W[k*ldw + n]
// Each wave owns a 64x32 block = 4(M) x 2(N) subtiles of 16x16, accumulated with
// V_WMMA_F32_16X16X4_F32.  Per K-step of 4: 4 A-loads + 2 B-loads feed 8 WMMAs
// (B reused 4x, A reused 2x out of registers -> ~10.7 FLOP/B from cache).
// A-matrix lanes 0-15 carry K={0,1}, lanes 16-31 K={2,3}; B mirrors with N on lanes.
struct GemmParams {
  const float* A; const float* W; float* C;
  const float* bias; const float* add;
  long long lda, ldw, ldc, ldadd;
  int M, N, K;
  float alpha;
  int flags;
  int transB;
  int innerB;                      // batch split: z -> (z/innerB, z%innerB)
  long long aO, aI, wO, wI, cO, cI;
};

#define MI 4
#define NI 2

__global__ __launch_bounds__(128) void gemm_f32_wmma(GemmParams p) {
  const int lane = threadIdx.x & 31;
  const int wib  = threadIdx.x >> 5;
  const int tilesN = (p.N + 31) >> 5;
  const int tilesM = (p.M + 63) >> 6;
  const int wave = blockIdx.x * 4 + wib;
  if (wave >= tilesM * tilesN) return;           // wave-uniform exit, EXEC stays all-1s
  const int tm = wave / tilesN;
  const int tn = wave % tilesN;

  const long long z  = blockIdx.z;
  const long long zq = z / p.innerB;
  const long long zr = z - zq * p.innerB;
  const long long aOff = zq * p.aO + zr * p.aI;
  const long long wOff = zq * p.wO + zr * p.wI;
  const long long cOff = zq * p.cO + zr * p.cI;

  const int half = lane >> 4;
  const int l15  = lane & 15;

  v8f acc[MI][NI];
#pragma unroll
  for (int mi = 0; mi < MI; ++mi)
#pragma unroll
    for (int ni = 0; ni < NI; ++ni)
      acc[mi][ni] = (v8f){0.f, 0.f, 0.f, 0.f, 0.f, 0.f, 0.f, 0.f};

  const float* Ap[MI];
#pragma unroll
  for (int mi = 0; mi < MI; ++mi) {
    int r = tm * 64 + mi * 16 + l15; if (r >= p.M) r = p.M - 1;   // clamp, no predication
    Ap[mi] = p.A + aOff + (long long)r * p.lda + (half << 1);
  }

#if defined(HAVE_WMMA_F32X4)
  if (!p.transB) {
    const float* Wp[NI];
#pragma unroll
    for (int ni = 0; ni < NI; ++ni) {
      int c = tn * 32 + ni * 16 + l15; if (c >= p.N) c = p.N - 1;
      Wp[ni] = p.W + wOff + (long long)c * p.ldw + (half << 1);
    }
    for (int k = 0; k < p.K; k += 4) {
      if ((k & 127) == 0) {                       // light-touch gfx1250 prefetch
        __builtin_prefetch(Ap[0] + k + 128, 0, 0);
        __builtin_prefetch(Wp[0] + k + 128, 0, 0);
      }
      v2f a[MI], b[NI];
#pragma unroll
      for (int mi = 0; mi < MI; ++mi) a[mi] = *(const v2f*)(Ap[mi] + k);
#pragma unroll
      for (int ni = 0; ni < NI; ++ni) b[ni] = *(const v2f*)(Wp[ni] + k);
#pragma unroll
      for (int mi = 0; mi < MI; ++mi)
#pragma unroll
        for (int ni = 0; ni < NI; ++ni)
          acc[mi][ni] = __builtin_amdgcn_wmma_f32_16x16x4_f32(
              false, a[mi], false, b[ni], (short)0, acc[mi][ni], false, false);
    }
  } else {
    const float* Wp[NI];
#pragma unroll
    for (int ni = 0; ni < NI; ++ni) {
      int c = tn * 32 + ni * 16 + l15; if (c >= p.N) c = p.N - 1;
      Wp[ni] = p.W + wOff + (long long)(half << 1) * p.ldw + c;
    }
    for (int k = 0; k < p.K; k += 4) {
      v2f a[MI], b[NI];
#pragma unroll
      for (int mi = 0; mi < MI; ++mi) a[mi] = *(const v2f*)(Ap[mi] + k);
#pragma unroll
      for (int ni = 0; ni < NI; ++ni) {
        b[ni].x = Wp[ni][(long long)k * p.ldw];
        b[ni].y = Wp[ni][(long long)(k + 1) * p.ldw];
      }
#pragma unroll
      for (int mi = 0; mi < MI; ++mi)
#pragma unroll
        for (int ni = 0; ni < NI; ++ni)
          acc[mi][ni] = __builtin_amdgcn_wmma_f32_16x16x4_f32(
              false, a[mi], false, b[ni], (short)0, acc[mi][ni], false, false);
    }
  }
#else
  { // VALU fallback producing the same WMMA C layout
    for (int k = 0; k < p.K; ++k) {
#pragma unroll
      for (int ni = 0; ni < NI; ++ni) {
        int cc = tn * 32 + ni * 16 + l15; if (cc >= p.N) cc = p.N - 1;
        float bv = p.transB ? p.W[wOff + (long long)k * p.ldw + cc]
                            : p.W[wOff + (long long)cc * p.ldw + k];
#pragma unroll
        for (int mi = 0; mi < MI; ++mi) {
          int rb = tm * 64 + mi * 16 + half * 8;
#pragma unroll
          for (int r = 0; r < 8; ++r) {
            int rr = rb + r; if (rr >= p.M) rr = p.M - 1;
            acc[mi][ni][r] += p.A[aOff + (long long)rr * p.lda + k] * bv;
          }
        }
      }
    }
  }
#endif

  // epilogue: C layout = VGPR r -> row (subtile base + half*8 + r), col (subtile base + l15)
#pragma unroll
  for (int ni = 0; ni < NI; ++ni) {
    const int ccol = tn * 32 + ni * 16 + l15;
    if (ccol < p.N) {
      const float bv = (p.flags & FLAG_BIAS) ? p.bias[ccol] : 0.f;
#pragma unroll
      for (int mi = 0; mi < MI; ++mi) {
        const int crowBase = tm * 64 + mi * 16 + half * 8;
#pragma unroll
        for (int r = 0; r < 8; ++r) {
          const int rr = crowBase + r;
          if (rr < p.M) {
            float x = acc[mi][ni][r];
            if (p.flags & FLAG_SCALE) x *= p.alpha;
            x += bv;
            if (p.flags & FLAG_RELU) x = fmaxf(x, 0.f);
            if (p.flags & FLAG_ADD)  x += p.add[(long long)rr * p.ldadd + ccol];
            p.C[cOff + (long long)rr * p.ldc + ccol] = x;
          }
        }
      }
    }
  }
}

// ================= helpers =================
__device__ __forceinline__ float bilin(const float* __restrict__ img, int H, int W, int C,
                                       int c, float x, float y) {
  float x0f = floorf(x), y0f = floorf(y);
  int x0 = (int)x0f, y0 = (int)y0f;
  int x1 = x0 + 1, y1 = y0 + 1;
  float fx = x - x0f, fy = y - y0f;
  bool vx0 = (x0 >= 0) && (x0 < W), vx1 = (x1 >= 0) && (x1 < W);
  bool vy0 = (y0 >= 0) && (y0 < H), vy1 = (y1 >= 0) && (y1 < H);
  int cx0 = x0 < 0 ? 0 : (x0 > W - 1 ? W - 1 : x0);
  int cx1 = x1 < 0 ? 0 : (x1 > W - 1 ? W - 1 : x1);
  int cy0 = y0 < 0 ? 0 : (y0 > H - 1 ? H - 1 : y0);
  int cy1 = y1 < 0 ? 0 : (y1 > H - 1 ? H - 1 : y1);
  float v00 = img[((long long)cy0 * W + cx0) * C + c];
  float v10 = img[((long long)cy0 * W + cx1) * C + c];
  float v01 = img[((long long)cy1 * W + cx0) * C + c];
  float v11 = img[((long long)cy1 * W + cx1) * C + c];
  return v00 * ((vx0 && vy0) ? (1.f - fx) * (1.f - fy) : 0.f)
       + v10 * ((vx1 && vy0) ? fx * (1.f - fy) : 0.f)
       + v01 * ((vx0 && vy1) ? (1.f - fx) * fy : 0.f)
       + v11 * ((vx1 && vy1) ? fx * fy : 0.f);
}

// channel adjust: BCHW -> BHWC 1x1 conv (gather-shaped, VALU)
__global__ __launch_bounds__(256) void adj_kernel(const float* __restrict__ feat,
                                                  const float* __restrict__ ca_w,
                                                  const float* __restrict__ ca_b,
                                                  float* __restrict__ out, int HW) {
  long long idx = (long long)blockIdx.x * blockDim.x + threadIdx.x;
  long long total = (long long)kB * HW * kC;
  if (idx >= total) return;
  int o = (int)(idx % kC);
  long long rem = idx / kC;
  int hw = (int)(rem % HW);
  int b = (int)(rem / HW);
  const float* f = feat + (long long)b * kCin * HW + hw;
  const float* w = ca_w + o * kCin;
  float s = ca_b[o];
#pragma unroll 8
  for (int c = 0; c < kCin; ++c) s += f[(long long)c * HW] * w[c];
  out[idx] = s;
}

// align_corners bilinear resize of the offset field
__global__ __launch_bounds__(256) void resize_kernel(const float* __restrict__ src,
                                                     float* __restrict__ dst,
                                                     int H, int W, int Ht, int Wt) {
  long long idx = (long long)blockIdx.x * blockDim.x + threadIdx.x;
  long long total = (long long)kB * Ht * Wt * kC;
  if (idx >= total) return;
  int c = (int)(idx % kC);
  long long r = idx / kC;
  int x = (int)(r % Wt); r /= Wt;
  int y = (int)(r % Ht);
  int b = (int)(r / Ht);
  float fy = (float)y * (float)(H - 1) / (float)(Ht - 1);
  float fx = (float)x * (float)(W - 1) / (float)(Wt - 1);
  dst[idx] = bilin(src + (long long)b * H * W * kC, H, W, kC, c, fx, fy);
}

// deformable sampling, fused with z-softmax level weighting:
// comb[b,p,s,:] (level==0 ? = : +=) zw[s,level] * bilinear(feats_l, px+dx, py+dy)
// one wave per (b,p,s); 72 channels strided over 32 lanes
__global__ __launch_bounds__(256) void sample_kernel(const float* __restrict__ feats,
                                                     const float* __restrict__ offs,
                                                     const float* __restrict__ priors,
                                                     const float* __restrict__ z_emb,
                                                     float* __restrict__ comb,
                                                     int Hl, int Wl, int level) {
  const int wid = (int)(((long long)blockIdx.x * blockDim.x + threadIdx.x) >> 5);
  const int lane = threadIdx.x & 31;
  if (wid >= kB * kP * kS) return;
  const int s = wid % kS;
  const int bp = wid / kS;
  const int b = bp / kP;

  const int sf = (kS - 1) - s;                                 // flip along s
  const int strip = (int)floorf(((float)sf / (float)(kS - 1)) * (float)kNS);
  const float pfx = priors[(long long)bp * kPr + 4 + strip];
  const float pfy = 1.f - (float)strip / (float)kNS;
  float u = fminf(fmaxf(pfx / (float)kNS, 0.f), 1.f);
  const float px = u * (float)(Wl - 1);
  const float py = pfy * (float)(Hl - 1);

  const float* offsB = offs + (long long)b * Hl * Wl * kC;
  const float dyv = bilin(offsB, Hl, Wl, kC, 2 * s, px, py);     // only channels 2s, 2s+1 used
  const float dxv = bilin(offsB, Hl, Wl, kC, 2 * s + 1, px, py);
  const float sx = px + dxv, sy = py + dyv;

  const float z = z_emb[s];
  const float e0 = __expf(-0.5f * z * z);
  const float e1 = __expf(-0.5f * (z - 1.f) * (z - 1.f));
  const float e2 = __expf(-0.5f * (z - 2.f) * (z - 2.f));
  const float zw = (level == 0 ? e0 : level == 1 ? e1 : e2) / (e0 + e1 + e2);

  const float* featB = feats + (long long)b * Hl * Wl * kC;
  float* dst = comb + (long long)wid * kC;

  float x0f = floorf(sx), y0f = floorf(sy);
  int x0 = (int)x0f, y0 = (int)y0f, x1 = x0 + 1, y1 = y0 + 1;
  float fx = sx - x0f, fy = sy - y0f;
  bool vx0 = (x0 >= 0) && (x0 < Wl), vx1 = (x1 >= 0) && (x1 < Wl);
  bool vy0 = (y0 >= 0) && (y0 < Hl), vy1 = (y1 >= 0) && (y1 < Hl);
  float m00 = (vx0 && vy0) ? (1.f - fx) * (1.f - fy) : 0.f;
  float m10 = (vx1 && vy0) ? fx * (1.f - fy) : 0.f;
  float m01 = (vx0 && vy1) ? (1.f - fx) * fy : 0.f;
  float m11 = (vx1 && vy1) ? fx * fy : 0.f;
  int cx0 = x0 < 0 ? 0 : (x0 > Wl - 1 ? Wl - 1 : x0);
  int cx1 = x1 < 0 ? 0 : (x1 > Wl - 1 ? Wl - 1 : x1);
  int cy0 = y0 < 0 ? 0 : (y0 > Hl - 1 ? Hl - 1 : y0);
  int cy1 = y1 < 0 ? 0 : (y1 > Hl - 1 ? Hl - 1 : y1);
  const float* p00 = featB + ((long long)cy0 * Wl + cx0) * kC;
  const float* p10 = featB + ((long long)cy0 * Wl + cx1) * kC;
  const float* p01 = featB + ((long long)cy1 * Wl + cx0) * kC;
  const float* p11 = featB + ((long long)cy1 * Wl + cx1) * kC;
  for (int c = lane; c < kC; c += 32) {
    float v = m00 * p00[c] + m10 * p10[c] + m01 * p01[c] + m11 * p11[c];
    if (level == 0) dst[c] = zw * v;
    else            dst[c] += zw * v;
  }
}

// fused bias for dp-folded gf: bf[d] = gf_b[d] + sum_{i,k} gf_w[d,i,k]*dp_b[k]
__global__ void fuse_bias_kernel(const float* __restrict__ gf_w, const float* __restrict__ gf_b,
                                 const float* __restrict__ dp_b, float* __restrict__ bf) {
  int d = blockIdx.x * blockDim.x + threadIdx.x;
  if (d >= kD) return;
  float s = gf_b[d];
  const float* w = gf_w + (long long)d * 9 * kC;
  for (int i = 0; i < 9; ++i)
    for (int k = 0; k < kC; ++k) s += w[i * kC + k] * dp_b[k];
  bf[d] = s;
}

// row softmax (wave per row, wave32 xor-shuffle reductions)
__global__ __launch_bounds__(256) void softmax_kernel(float* __restrict__ scores, int rows, int cols) {
  int row = (int)(((long long)blockIdx.x * blockDim.x + threadIdx.x) >> 5);
  int lane = threadIdx.x & 31;
  if (row >= rows) return;
  float* r = scores + (long long)row * cols;
  float m = -3.4e38f;
  for (int i = lane; i < cols; i += 32) m = fmaxf(m, r[i]);
#pragma unroll
  for (int o = 16; o > 0; o >>= 1) m = fmaxf(m, __shfl_xor(m, o, 32));
  float s = 0.f;
  for (int i = lane; i < cols; i += 32) { float e = __expf(r[i] - m); r[i] = e; s += e; }
#pragma unroll
  for (int o = 16; o > 0; o >>= 1) s += __shfl_xor(s, o, 32);
  float inv = 1.f / s;
  for (int i = lane; i < cols; i += 32) r[i] *= inv;
}

// scale tail coeffs, write coeffs + sigmoid(poly basis) into the output tensor
__global__ __launch_bounds__(256) void finalize_kernel(const float* __restrict__ coef,
                                                       const float* __restrict__ b_scale,
                                                       const float* __restrict__ b_bias,
                                                       float* __restrict__ out) {
  long long idx = (long long)blockIdx.x * blockDim.x + threadIdx.x;
  const long long total = (long long)kB * kP * 72;
  if (idx >= total) return;
  int o = (int)(idx % 72);
  long long n = idx / 72;
  const float sc = b_scale[0], sb = b_bias[0];
  float c0 = coef[n * 4 + 0];
  float c1 = coef[n * 4 + 1];
  float c2 = coef[n * 4 + 2] * sc + sb;
  float c3 = coef[n * 4 + 3] * sc + sb;
  float* orow = out + n * kOut;
  if (o == 0) { orow[2] = c0; orow[3] = c1; orow[4] = c2; orow[5] = c3; }
  float y = (float)o / 71.f;
  float y2 = y * y;
  float v = c0 + c1 * y + c2 * y2 + c3 * y2 * y;
  orow[6 + o] = 1.f / (1.f + __expf(-v));
}

// ================= host-side launch helper =================
static void run_gemm(hipStream_t stream,
                     const float* A, long long lda, const float* W, long long ldw, int transB,
                     float* C, long long ldc, const float* bias, const float* add, long long ldadd,
                     int M, int N, int K, float alpha, int flags,
                     int batch = 1, int innerB = 1,
                     long long aO = 0, long long aI = 0, long long wO = 0, long long wI = 0,
                     long long cO = 0, long long cI = 0) {
  GemmParams p;
  p.A = A; p.W = W; p.C = C; p.bias = bias; p.add = add;
  p.lda = lda; p.ldw = ldw; p.ldc = ldc; p.ldadd = ldadd;
  p.M = M; p.N = N; p.K = K; p.alpha = alpha; p.flags = flags; p.transB = transB;
  p.innerB = innerB; p.aO = aO; p.aI = aI; p.wO = wO; p.wI = wI; p.cO = cO; p.cI = cI;
  int waves = ((M + 63) / 64) * ((N + 31) / 32);
  dim3 grid((unsigned)((waves + 3) / 4), 1, (unsigned)batch);
  gemm_f32_wmma<<<grid, dim3(128, 1, 1), 0, stream>>>(p);
}

extern "C" void kernel_launch(void* const* d_in, const int* in_sizes, int n_in,
                              void* d_out, int out_size, void* d_ws, size_t ws_size,
                              hipStream_t stream) {
  const float* feat0  = (const float*)d_in[0];
  const float* feat1  = (const float*)d_in[1];
  const float* feat2  = (const float*)d_in[2];
  const float* priors = (const float*)d_in[3];
  const float* ca_w   = (const float*)d_in[4];
  const float* ca_b   = (const float*)d_in[5];
  const float* off_w  = (const float*)d_in[6];
  const float* off_b  = (const float*)d_in[7];
  const float* dp_w   = (const float*)d_in[8];
  const float* dp_b   = (const float*)d_in[9];
  const float* z_emb  = (const float*)d_in[10];
  const float* gf_w   = (const float*)d_in[11];
  const float* gf_b   = (const float*)d_in[12];
  const float* wq = (const float*)d_in[13]; const float* bq = (const float*)d_in[14];
  const float* wk = (const float*)d_in[15]; const float* bk = (const float*)d_in[16];
  const float* wv = (const float*)d_in[17]; const float* bv = (const float*)d_in[18];
  const float* wo = (const float*)d_in[19]; const float* bo = (const float*)d_in[20];
  const float* cf_w1 = (const float*)d_in[21]; const float* cf_b1 = (const float*)d_in[22];
  const float* cf_w2 = (const float*)d_in[23]; const float* cf_b2 = (const float*)d_in[24];
  const float* cls_w1 = (const float*)d_in[25]; const float* cls_b1 = (const float*)d_in[26];
  const float* cls_w2 = (const float*)d_in[27]; const float* cls_b2 = (const float*)d_in[28];
  const float* reg_w1 = (const float*)d_in[29]; const float* reg_b1 = (const float*)d_in[30];
  const float* p_w1 = (const float*)d_in[31]; const float* p_b1 = (const float*)d_in[32];
  const float* p_w2 = (const float*)d_in[33]; const float* p_b2 = (const float*)d_in[34];
  const float* b_scale = (const float*)d_in[35];
  const float* b_bias  = (const float*)d_in[36];

  float* ws  = (float*)d_ws;
  float* out = (float*)d_out;
  const int BP = kB * kP;                      // 6144

  // ---- 1. channel adjust for three pyramid levels (BCHW -> BHWC) ----
  adj_kernel<<<(2304000 + 255) / 256, 256, 0, stream>>>(feat0, ca_w, ca_b, ws + OFF_FEATS0, 4000);
  adj_kernel<<<( 576000 + 255) / 256, 256, 0, stream>>>(feat1, ca_w, ca_b, ws + OFF_FEATS1, 1000);
  adj_kernel<<<( 144000 + 255) / 256, 256, 0, stream>>>(feat2, ca_w, ca_b, ws + OFF_FEATS2,  250);

  // ---- 2. offset field on level0: [32000,72] x [72,72]^T + b  (WMMA) ----
  run_gemm(stream, ws + OFF_FEATS0, kC, off_w, kC, 0, ws + OFF_OFFS0, kC,
           off_b, nullptr, 0, 32000, kC, kC, 1.f, FLAG_BIAS);

  // ---- 3. align-corners resize of offsets to levels 1 & 2 ----
  resize_kernel<<<(576000 + 255) / 256, 256, 0, stream>>>(ws + OFF_OFFS0, ws + OFF_OFFS1, 40, 100, 20, 50);
  resize_kernel<<<(144000 + 255) / 256, 256, 0, stream>>>(ws + OFF_OFFS0, ws + OFF_OFFS2, 40, 100, 10, 25);

  // ---- 4. deformable sampling fused with z-softmax weighting -> comb[b,p,s,72] ----
  const int sampBlocks = (kB * kP * kS) / 8;   // wave per (b,p,s), 8 waves/block
  sample_kernel<<<sampBlocks, 256, 0, stream>>>(ws + OFF_FEATS0, ws + OFF_OFFS0, priors, z_emb, ws + OFF_COMB, 40, 100, 0);
  sample_kernel<<<sampBlocks, 256, 0, stream>>>(ws + OFF_FEATS1, ws + OFF_OFFS1, priors, z_emb, ws + OFF_COMB, 20,  50, 1);
  sample_kernel<<<sampBlocks, 256, 0, stream>>>(ws + OFF_FEATS2, ws + OFF_OFFS2, priors, z_emb, ws + OFF_COMB, 10,  25, 2);

  // ---- 5. fold dp into gf: Wf[4608,72] = gf_w[4608,72] @ dp_w[72,72]  (transB WMMA) ----
  run_gemm(stream, gf_w, kC, dp_w, kC, 1, ws + OFF_WF, kC,
           nullptr, nullptr, 0, kD * 9, kC, kC, 1.f, 0);
  fuse_bias_kernel<<<2, 256, 0, stream>>>(gf_w, gf_b, dp_b, ws + OFF_BF);

  // ---- 6. grouped gf (4 GEMMs): feature[6144,512] ----
  for (int g = 0; g < 4; ++g) {
    run_gemm(stream, ws + OFF_COMB + g * 648, kS * kC,          // A slice: cols [648g,648g+648)
             ws + OFF_WF + (long long)g * 128 * 648, 648, 0,    // W rows o: 648 contiguous
             ws + OFF_FEAT + g * 128, kD,
             ws + OFF_BF + g * 128, nullptr, 0,
             BP, 128, 648, 1.f, FLAG_BIAS);
  }

  // ---- 7. Q, K, V projections ----
  run_gemm(stream, ws + OFF_FEAT, kD, wq, kD, 0, ws + OFF_Q, kD, bq, nullptr, 0, BP, kD, kD, 1.f, FLAG_BIAS);
  run_gemm(stream, ws + OFF_FEAT, kD, wk, kD, 0, ws + OFF_K, kD, bk, nullptr, 0, BP, kD, kD, 1.f, FLAG_BIAS);
  run_gemm(stream, ws + OFF_FEAT, kD, wv, kD, 0, ws + OFF_V, kD, bv, nullptr, 0, BP, kD, kD, 1.f, FLAG_BIAS);

  // ---- 8. scores[b,g] = (Q_bg @ K_bg^T) / sqrt(128), batched over z = b*4+g ----
  run_gemm(stream, ws + OFF_Q, kD, ws + OFF_K, kD, 0, ws + OFF_SCORES, kP,
           nullptr, nullptr, 0, kP, kP, 128, 1.f / sqrtf(128.f), FLAG_SCALE,
           32, 4,
           (long long)kP * kD, 128,            // A: per-b, per-g
           (long long)kP * kD, 128,            // W(=K): per-b, per-g
           (long long)4 * kP * kP, (long long)kP * kP);

  // ---- 9. row softmax over 32*768 rows ----
  softmax_kernel<<<(32 * kP) / 8, 256, 0, stream>>>(ws + OFF_SCORES, 32 * kP, kP);

  // ---- 10. res[b,g] = attn @ V_bg  (transB: B[k,n] = v[b,k,g*128+n]) ----
  run_gemm(stream, ws + OFF_SCORES, kP, ws + OFF_V, kD, 1, ws + OFF_RES, kD,
           nullptr, nullptr, 0, kP, 128, kP, 1.f, 0,
           32, 4,
           (long long)4 * kP * kP, (long long)kP * kP,
           (long long)kP * kD, 128,
           (long long)kP * kD, 128);

  // ---- 11. output projection + FFN with residual ----
  run_gemm(stream, ws + OFF_RES, kD, wo, kD, 0, ws + OFF_RES2, kD, bo, nullptr, 0, BP, kD, kD, 1.f, FLAG_BIAS);
  run_gemm(stream, ws + OFF_RES2, kD, cf_w1, kD, 0, ws + OFF_T1, 2 * kD, cf_b1, nullptr, 0,
           BP, 2 * kD, kD, 1.f, FLAG_BIAS | FLAG_RELU);
  run_gemm(stream, ws + OFF_T1, 2 * kD, cf_w2, 2 * kD, 0, ws + OFF_FEAT2, kD, cf_b2,
           ws + OFF_FEAT, kD, BP, kD, 2 * kD, 1.f, FLAG_BIAS | FLAG_ADD);

  // ---- 12. classification head -> out cols [0,2) ----
  run_gemm(stream, ws + OFF_FEAT2, kD, cls_w1, kD, 0, ws + OFF_CLSH, kD, cls_b1, nullptr, 0,
           BP, kD, kD, 1.f, FLAG_BIAS | FLAG_RELU);
  run_gemm(stream, ws + OFF_CLSH, kD, cls_w2, kD, 0, out, kOut, cls_b2, nullptr, 0,
           BP, 2, kD, 1.f, FLAG_BIAS);

  // ---- 13. regression + polynomial head ----
  run_gemm(stream, ws + OFF_FEAT2, kD, reg_w1, kD, 0, ws + OFF_REG, kD, reg_b1, nullptr, 0,
           BP, kD, kD, 1.f, FLAG_BIAS | FLAG_RELU);
  run_gemm(stream, ws + OFF_REG, kD, p_w1, kD, 0, ws + OFF_PH, 2 * kD, p_b1, nullptr, 0,
           BP, 2 * kD, kD, 1.f, FLAG_BIAS | FLAG_RELU);
  run_gemm(stream, ws + OFF_PH, 2 * kD, p_w2, 2 * kD, 0, ws + OFF_COEF, 4, p_b2, nullptr, 0,
           BP, 4, 2 * kD, 1.f, FLAG_BIAS);

  // ---- 14. coeff scaling + sigmoid(poly) -> out cols [2,78) ----
  finalize_kernel<<<(BP * 72 + 255) / 256, 256, 0, stream>>>(ws + OFF_COEF, b_scale, b_bias, out);

  (void)in_sizes; (void)n_in; (void)out_size; (void)ws_size;
}